// SelectiveScanSplit_6828998001047
// MI455X (gfx1250) — compile-verified
//
#include <hip/hip_runtime.h>
#include <math.h>

typedef float v2f __attribute__((ext_vector_type(2)));
typedef float v8f __attribute__((ext_vector_type(8)));

// Problem constants (from reference setup_inputs)
constexpr int Lw  = 2048;   // sequence length
constexpr int Nst = 16;     // state dim
constexpr int CH  = 16;     // chunk (timesteps per WMMA tile)

// Merge-reduce 8 registers over 16-lane segments:
// returns, in every lane l, sum over the segment of v[l & 7].
__device__ __forceinline__ float merge_reduce8(const float v[8], int lane)
{
    float a0 = v[0] + __shfl_xor(v[0], 1, 16);
    float a1 = v[1] + __shfl_xor(v[1], 1, 16);
    float a2 = v[2] + __shfl_xor(v[2], 1, 16);
    float a3 = v[3] + __shfl_xor(v[3], 1, 16);
    float a4 = v[4] + __shfl_xor(v[4], 1, 16);
    float a5 = v[5] + __shfl_xor(v[5], 1, 16);
    float a6 = v[6] + __shfl_xor(v[6], 1, 16);
    float a7 = v[7] + __shfl_xor(v[7], 1, 16);
    float w0 = (lane & 1) ? a1 : a0;      // reg bit0 <- lane bit0
    float w1 = (lane & 1) ? a3 : a2;
    float w2 = (lane & 1) ? a5 : a4;
    float w3 = (lane & 1) ? a7 : a6;
    float x0 = w0 + __shfl_xor(w0, 2, 16);
    float x1 = w1 + __shfl_xor(w1, 2, 16);
    float x2 = w2 + __shfl_xor(w2, 2, 16);
    float x3 = w3 + __shfl_xor(w3, 2, 16);
    float y0 = (lane & 2) ? x1 : x0;      // reg bit1 <- lane bit1
    float y1 = (lane & 2) ? x3 : x2;
    float z0 = y0 + __shfl_xor(y0, 4, 16);
    float z1 = y1 + __shfl_xor(y1, 4, 16);
    float f  = (lane & 4) ? z1 : z0;      // reg bit2 <- lane bit2
    f += __shfl_xor(f, 8, 16);
    return f;
}

// One wave (32 lanes) owns one (b,d) row. Chunked selective scan:
//   h_t = exp(A*cs_t) * h_{-1} + sum_{s<=t} exp(A*(cs_t-cs_s)) B_s x_s
//   y_t = C_t . h_t
// With C'[n,t]=C*exp(A cs_t), B''[n,s]=B*x_s*exp(-A cs_s):
//   M = C'^T B''  (16x16x16 matmul -> 4x V_WMMA_F32_16X16X4_F32)
//   y = tril(M) row-sums + C'^T h_prev ; h_next = e_end*(h_prev + colsum(B''))
__global__ __launch_bounds__(128)
void selscan_wmma_kernel(const float* __restrict__ u,
                         const float* __restrict__ delta,
                         const float* __restrict__ Am,
                         const float* __restrict__ Bm,
                         const float* __restrict__ Cm,
                         const float* __restrict__ Dv,
                         const float* __restrict__ z,
                         float* __restrict__ out)
{
    const int wave = threadIdx.x >> 5;
    const int lane = threadIdx.x & 31;
    const int w    = blockIdx.x * 4 + wave;   // 0..2047 -> (b,d)
    const int b    = w >> 10;
    const int d    = w & 1023;

    const int hi  = lane >> 4;    // half: 0 = lanes 0-15, 1 = lanes 16-31
    const int t16 = lane & 15;    // t (and s) index within chunk
    const int q   = lane & 7;     // register class in merged layouts
    const int nh  = 4 * (q >> 1) + (q & 1) + 2 * hi;  // n owned by this lane's h slot
    const bool selHalf = ((t16 >> 3) == hi);          // y gather: own half holds lane's t?
    (void)nh;

    const float* urow = u     + (size_t)w * Lw;
    const float* drow = delta + (size_t)w * Lw;
    const float* zrow = z     + (size_t)w * Lw;
    float*       orow = out   + (size_t)w * Lw;
    const float* Bb = Bm + (size_t)b * Nst * Lw;
    const float* Cb = Cm + (size_t)b * Nst * Lw;

    // Per-lane A values for the K-slots this lane feeds: n = 4j + v + 2*hi
    float An[4][2];
#pragma unroll
    for (int j = 0; j < 4; ++j) {
        An[j][0] = Am[d * Nst + 4 * j + 0 + 2 * hi];
        An[j][1] = Am[d * Nst + 4 * j + 1 + 2 * hi];
    }
    const float Dsc = Dv[d];

    // tril masks per D-matrix register r: keep s<=t, t = r + 8*hi, s = t16
    bool keep[8];
#pragma unroll
    for (int r = 0; r < 8; ++r) keep[r] = (t16 <= r + 8 * hi);

    // State h lives in a register, merged layout: lane holds h[nh] (dup in lanes 8..15)
    float hreg = 0.0f;

    for (int t0 = 0; t0 < Lw; t0 += CH) {
        // ---- per-timestep scalars, replicated across both halves (t = t16) ----
        float dl = drow[t0 + t16];
        float ul = urow[t0 + t16];
        float zl = zrow[t0 + t16];
        float sp = (dl > 20.f) ? dl : log1pf(__expf(dl));   // softplus(delta)
        float x  = sp * ul;                                 // delta*u

        // inclusive prefix sum of sp within 16-lane segments -> cs[t]
        float cs = sp;
#pragma unroll
        for (int o = 1; o < 16; o <<= 1) {
            float tv = __shfl_up(cs, o, 16);
            if (t16 >= o) cs += tv;
        }

        v8f acc = {0.f, 0.f, 0.f, 0.f, 0.f, 0.f, 0.f, 0.f};
        float yin = 0.f;
        float bsum[8];
        float eend[8];

#pragma unroll
        for (int j = 0; j < 4; ++j) {
            const int n0 = 4 * j + 2 * hi;       // K-slot v=0
            const int n1 = n0 + 1;               // K-slot v=1
            // exp(A*cs), clamped so 1/e stays finite in fp32
            float e0 = __expf(fmaxf(An[j][0] * cs, -80.f));
            float e1 = __expf(fmaxf(An[j][1] * cs, -80.f));
            // A-operand: C'[n,t]
            float a0 = Cb[(size_t)n0 * Lw + t0 + t16] * e0;
            float a1 = Cb[(size_t)n1 * Lw + t0 + t16] * e1;
            // initial-state term: C'[n,t]*h_prev[n]; h broadcast from merged reg
            float h0 = __shfl(hreg, 2 * j,     16);   // lane 2j   of segment: n=4j+2hi
            float h1 = __shfl(hreg, 2 * j + 1, 16);   // lane 2j+1 of segment: n=4j+1+2hi
            yin += a0 * h0 + a1 * h1;
            // B-operand: B''[n,s] = B * x_s * exp(-A*cs_s)
            float r0 = __builtin_amdgcn_rcpf(e0);
            float r1 = __builtin_amdgcn_rcpf(e1);
            float b0 = Bb[(size_t)n0 * Lw + t0 + t16] * r0 * x;
            float b1 = Bb[(size_t)n1 * Lw + t0 + t16] * r1 * x;

            v2f av = {a0, a1};
            v2f bv = {b0, b1};
            acc = __builtin_amdgcn_wmma_f32_16x16x4_f32(
                false, av, false, bv, (short)0, acc, false, false);

            bsum[2 * j + 0] = b0;
            bsum[2 * j + 1] = b1;
            // e_end[n] = exp(A*cs[15]) : grab lane 15 of each segment
            eend[2 * j + 0] = __shfl(e0, 15, 16);
            eend[2 * j + 1] = __shfl(e1, 15, 16);
        }

        // ---- tril mask + merged row-sum of M'' ----
        float mreg[8];
#pragma unroll
        for (int r = 0; r < 8; ++r) mreg[r] = keep[r] ? acc[r] : 0.f;
        float ymerged = merge_reduce8(mreg, lane);        // lane: Ysum[t = (lane&7)+8*hi]
        float yother  = __shfl_xor(ymerged, 16, 32);      // other half's set
        float ymat    = selHalf ? ymerged : yother;       // lane now holds Ysum[t16]

        // ---- h update: h_next = e_end * (h_prev + colsum_s B''[n,s]) ----
        float S = merge_reduce8(bsum, lane);              // lane: colsum for n = nh
        float ew0 = (lane & 1) ? eend[1] : eend[0];
        float ew1 = (lane & 1) ? eend[3] : eend[2];
        float ew2 = (lane & 1) ? eend[5] : eend[4];
        float ew3 = (lane & 1) ? eend[7] : eend[6];
        float ex0 = (lane & 2) ? ew1 : ew0;
        float ex1 = (lane & 2) ? ew3 : ew2;
        float e_m = (lane & 4) ? ex1 : ex0;               // lane: e_end[nh]
        hreg = e_m * (hreg + S);                          // uniform, no divergence

        // ---- finalize: y = (M-part + init-part + u*D) * silu(z) ----
        yin += __shfl_xor(yin, 16, 32);          // combine half-n partial sums
        float yv  = fmaf(ul, Dsc, ymat + yin);
        float sig = __builtin_amdgcn_rcpf(1.f + __expf(-zl));
        yv *= zl * sig;                          // silu(z) = z*sigmoid(z)
        if (lane < 16) orow[t0 + t16] = yv;

        // prefetch next chunk (gfx1250 global_prefetch_b8)
        if (t0 + CH < Lw) {
            __builtin_prefetch(drow + t0 + CH, 0, 0);
            __builtin_prefetch(urow + t0 + CH, 0, 0);
            __builtin_prefetch(zrow + t0 + CH, 0, 0);
        }
    }
}

extern "C" void kernel_launch(void* const* d_in, const int* in_sizes, int n_in,
                              void* d_out, int out_size, void* d_ws, size_t ws_size,
                              hipStream_t stream) {
    (void)in_sizes; (void)n_in; (void)d_ws; (void)ws_size; (void)out_size;
    const float* u     = (const float*)d_in[0];
    const float* delta = (const float*)d_in[1];
    const float* A     = (const float*)d_in[2];
    const float* B     = (const float*)d_in[3];
    const float* C     = (const float*)d_in[4];
    const float* D     = (const float*)d_in[5];
    const float* z     = (const float*)d_in[6];
    // d_in[7] = delta_softplus flag (asserted true in reference)
    float* out = (float*)d_out;

    dim3 grid(512);    // 2048 waves, one per (b,d) row
    dim3 block(128);   // 4 waves per workgroup
    hipLaunchKernelGGL(selscan_wmma_kernel, grid, block, 0, stream,
                       u, delta, A, B, C, D, z, out);
}